// DotProductAttention_80049600462927
// MI455X (gfx1250) — compile-verified
//
#include <hip/hip_runtime.h>

// MI455X / gfx1250, wave32. fp32 attention-weights (QK^T -> causal softmax).
// Output-bandwidth bound: 512MB out @ 23.3TB/s ~ 23us floor; ~0.009 TFLOP of
// matmul is negligible, so fp32 V_WMMA_F32_16X16X4_F32 is the right op.

typedef __attribute__((ext_vector_type(2))) float v2f;
typedef __attribute__((ext_vector_type(4))) float v4f;
typedef __attribute__((ext_vector_type(8))) float v8f;

#define B_ 2
#define S_ 2048
#define H_ 16
#define D_ 64
#define QT 16                   // q-rows per workgroup
#define NW 8                    // waves per workgroup (256 threads, wave32)
#define LDS_STRIDE (S_ + 4)     // pad 4 floats: rows 8 apart land 32 banks apart
#define NTILES (S_ / 16)        // 128 column tiles of 16

__global__ __launch_bounds__(256)
void attn_weights_wmma_f32(const float* __restrict__ Q,
                           const float* __restrict__ K,
                           float* __restrict__ Out) {
  extern __shared__ float smem[];          // [QT][LDS_STRIDE] raw scores

  const int qt = blockIdx.x & (NTILES - 1);        // q-tile index 0..127
  const int h  = (blockIdx.x >> 7) & (H_ - 1);
  const int b  = blockIdx.x >> 11;
  const int q0 = qt * QT;

  const int lane = threadIdx.x & 31;
  const int wave = threadIdx.x >> 5;

  // ISA 7.12.2: f32 16x4 A-matrix -- lanes 0..15 hold M=lane, K={0,1};
  // lanes 16..31 hold M=lane-16, K={2,3}. B (4x16) mirrors with N=lane&15.
  const int m    = lane & 15;
  const int koff = (lane >> 4) << 1;               // 0 or 2

  // ---------------- load A fragments: Q tile (16x64), pre-scaled 1/sqrt(D) ---
  const float* qrow = Q + ((size_t)((b * S_ + q0 + m) * H_ + h)) * D_;
  v2f a[16];
#pragma unroll
  for (int kk = 0; kk < 16; ++kk) {
    const float* p = qrow + kk * 4 + koff;
    a[kk].x = p[0] * 0.125f;
    a[kk].y = p[1] * 0.125f;
  }

  // ---------------- score phase: wave w owns tiles g = w, w+8, w+16, ... -----
  const float* kbase = K + ((size_t)(b * S_) * H_ + h) * D_;
#pragma unroll 1
  for (int t = 0; t < NTILES / NW; ++t) {
    const int g  = wave + t * NW;                  // global column-tile index
    const int k0 = g * 16;
    if (k0 > q0 + 15) break;                       // fully masked from here on

    const float* krow = kbase + (size_t)(k0 + m) * (H_ * D_);
    v8f acc = {0.f, 0.f, 0.f, 0.f, 0.f, 0.f, 0.f, 0.f};
#pragma unroll
    for (int kk = 0; kk < 16; ++kk) {
      v2f bf;
      const float* p = krow + kk * 4 + koff;
      bf.x = p[0];
      bf.y = p[1];
      // D = A(16x4) x B(4x16) + C ; chained accumulate over D_=64
      acc = __builtin_amdgcn_wmma_f32_16x16x4_f32(false, a[kk], false, bf,
                                                  (short)0, acc, false, false);
    }
    // C/D layout: vgpr v, lane l -> row v + (l>=16 ? 8 : 0), col = k0 + (l&15)
    const int rbase = (lane >> 4) << 3;
    const int col   = k0 + m;
#pragma unroll
    for (int v = 0; v < 8; ++v)
      smem[(rbase + v) * LDS_STRIDE + col] = acc[v];
  }

  __syncthreads();

  // ---------------- softmax + streamed write: wave w owns rows w and w+8 -----
  float* outb = Out + ((size_t)(b * H_ + h) * S_ + q0) * S_;
#pragma unroll 1
  for (int rr = 0; rr < 2; ++rr) {
    const int r = wave + rr * NW;
    const int q = q0 + r;                          // valid cols: 0..q inclusive
    float* row  = smem + r * LDS_STRIDE;

    // pass 1: row max over valid columns
    float mx = -3.402823466e38f;
#pragma unroll 1
    for (int i = 0; i < S_ / 128; ++i) {
      const int base = i * 128 + lane * 4;
      if (base <= q) {
        v4f vv = *(const v4f*)(row + base);
        mx = fmaxf(mx, vv.x);
        if (base + 1 <= q) mx = fmaxf(mx, vv.y);
        if (base + 2 <= q) mx = fmaxf(mx, vv.z);
        if (base + 3 <= q) mx = fmaxf(mx, vv.w);
      }
    }
#pragma unroll
    for (int off = 16; off > 0; off >>= 1)
      mx = fmaxf(mx, __shfl_xor(mx, off, 32));

    // pass 2: exp once per valid element, write e back to LDS, accumulate sum
    float sum = 0.f;
#pragma unroll 1
    for (int i = 0; i < S_ / 128; ++i) {
      const int base = i * 128 + lane * 4;
      if (base <= q) {
        v4f vv = *(const v4f*)(row + base);
        v4f e;
        e.x = __expf(vv.x - mx);
        e.y = (base + 1 <= q) ? __expf(vv.y - mx) : 0.f;
        e.z = (base + 2 <= q) ? __expf(vv.z - mx) : 0.f;
        e.w = (base + 3 <= q) ? __expf(vv.w - mx) : 0.f;
        sum += e.x + e.y + e.z + e.w;
        *(v4f*)(row + base) = e;                   // own rows only: no race
      }
    }
#pragma unroll
    for (int off = 16; off > 0; off >>= 1)
      sum += __shfl_xor(sum, off, 32);
    const float inv = 1.0f / sum;

    // pass 3: normalize and stream out (non-temporal: never reread, keep L2 clean)
    float* orow = outb + (size_t)r * S_;
#pragma unroll 1
    for (int i = 0; i < S_ / 128; ++i) {
      const int base = i * 128 + lane * 4;
      v4f o = {0.f, 0.f, 0.f, 0.f};
      if (base <= q) {
        v4f e = *(const v4f*)(row + base);
        o.x = e.x * inv;
        o.y = e.y * inv;
        o.z = e.z * inv;
        o.w = e.w * inv;
      }
      __builtin_nontemporal_store(o, (v4f*)(orow + base));
    }
  }
}

extern "C" void kernel_launch(void* const* d_in, const int* in_sizes, int n_in,
                              void* d_out, int out_size, void* d_ws, size_t ws_size,
                              hipStream_t stream) {
  const float* Q   = (const float*)d_in[0];
  const float* Kp  = (const float*)d_in[1];
  // d_in[2] is the tril mask; causal masking is applied analytically.
  float* Out = (float*)d_out;

  dim3 grid(B_ * H_ * (S_ / QT));                  // 4096 workgroups
  dim3 block(256);                                 // 8 waves (wave32)
  size_t shmem = (size_t)QT * LDS_STRIDE * sizeof(float);  // ~128.25 KB / WGP ok
  hipLaunchKernelGGL(attn_weights_wmma_f32, grid, block, shmem, stream, Q, Kp, Out);
}